// FastGTLayer_40948218200526
// MI455X (gfx1250) — compile-verified
//
#include <hip/hip_runtime.h>
#include <math.h>

#define C_OUT 3
#define C_IN  4
#define DD    64   // feature dim

#define TILE_EDGES 16   // edges per stage (256 threads / 16 lanes-per-edge)
#define STAGES     10   // 160 edges per block; 1M edges -> 6250 blocks exactly

// ---------------------------------------------------------------------------
// f32 global atomic add -> hardware GLOBAL_ATOMIC_ADD_F32 (no-return).
// ---------------------------------------------------------------------------
__device__ __forceinline__ void atomic_fadd(float* p, float v) {
#if __has_builtin(__builtin_amdgcn_global_atomic_fadd_f32)
    __builtin_amdgcn_global_atomic_fadd_f32(p, v);
#else
    (void)unsafeAtomicAdd(p, v);
#endif
}

// ---------------------------------------------------------------------------
// Kernel 0: zero the Hs accumulator region of d_out (float4 stores).
// ---------------------------------------------------------------------------
__global__ void fgt_zero_kernel(float4* __restrict__ out4, int n4) {
    int i = blockIdx.x * blockDim.x + threadIdx.x;
    if (i < n4) out4[i] = make_float4(0.f, 0.f, 0.f, 0.f);
}

// ---------------------------------------------------------------------------
// Kernel 1: filt = softmax(weight, axis=1).  weight is [C_OUT, C_IN] = 3x4.
// ---------------------------------------------------------------------------
__global__ void fgt_softmax_kernel(const float* __restrict__ w,
                                   float* __restrict__ filt_ws,
                                   float* __restrict__ filt_out) {
    int i = threadIdx.x;
    if (i < C_OUT) {
        float x[C_IN];
        float m = -INFINITY;
#pragma unroll
        for (int t = 0; t < C_IN; ++t) {
            x[t] = w[i * C_IN + t];
            m = fmaxf(m, x[t]);
        }
        float s = 0.f;
#pragma unroll
        for (int t = 0; t < C_IN; ++t) {
            x[t] = expf(x[t] - m);
            s += x[t];
        }
        float inv = 1.f / s;
#pragma unroll
        for (int t = 0; t < C_IN; ++t) {
            float f = x[t] * inv;
            filt_ws[i * C_IN + t]  = f;
            filt_out[i * C_IN + t] = f;
        }
    }
}

// ---------------------------------------------------------------------------
// Kernel 2: scatter-add SpMM, software-pipelined through the CDNA5 async
// gather-to-LDS engine (GLOBAL_LOAD_ASYNC_TO_LDS_B128 / ASYNCcnt).
// ---------------------------------------------------------------------------
struct EdgeMeta {
    int    active;
    int    row;
    float  s0, s1, s2;
    size_t coff;   // col*DD + lane16*4 (element offset into each H channel)
};

__global__ void __launch_bounds__(256)
fgt_scatter_kernel(const float* __restrict__ H,      // [C_OUT, N, D]
                   const int*   __restrict__ EI,     // [C_IN, 2, E]
                   const float* __restrict__ EV,     // [C_IN, E]
                   const float* __restrict__ filt,   // [C_OUT, C_IN]
                   float*       __restrict__ out,    // [C_OUT, N, D]
                   int N, int E, int M) {            // M = C_IN * E
    // Double-buffered staging: [buf][thread][channel] float4 = 24 KB LDS.
    __shared__ float4 stage_buf[2][256][C_OUT];

    const int  tid    = threadIdx.x;
    const int  lane16 = tid & 15;    // float4 slot within the 64-float row
    const int  sub    = tid >> 4;    // edge slot within the stage (0..15)
    const long base   = (long)blockIdx.x * (TILE_EDGES * STAGES) + sub;

    // ---- stage-local helpers (lambdas keep everything in registers) -------
    auto load_meta = [&](int s) {
        EdgeMeta m;
        long e   = base + (long)s * TILE_EDGES;
        m.active = (e < M);
        if (m.active) {
            int t = (int)(e / E);
            int j = (int)(e - (long)t * E);
            const int* EIt = EI + (size_t)t * 2 * E;
            m.row     = __builtin_nontemporal_load(EIt + j);
            int col   = __builtin_nontemporal_load(EIt + E + j);
            float ev  = __builtin_nontemporal_load(EV + (size_t)t * E + j);
            m.s0 = filt[0 * C_IN + t] * ev;
            m.s1 = filt[1 * C_IN + t] * ev;
            m.s2 = filt[2 * C_IN + t] * ev;
            m.coff = (size_t)col * DD + (size_t)lane16 * 4;
        } else {
            m.row = 0; m.s0 = m.s1 = m.s2 = 0.f; m.coff = 0;
        }
        return m;
    };

    auto issue_async = [&](int buf, const EdgeMeta& m) {
        if (m.active) {
#pragma unroll
            for (int c = 0; c < C_OUT; ++c) {
                unsigned long long ga =
                    (unsigned long long)(const void*)(H + (size_t)c * N * DD + m.coff);
                // low 32 bits of the flat shared address == LDS byte offset
                unsigned la = (unsigned)(unsigned long long)&stage_buf[buf][tid][c];
                asm volatile("global_load_async_to_lds_b128 %0, %1, off"
                             :: "v"(la), "v"(ga)
                             : "memory");
            }
        }
    };

    auto consume = [&](int buf, const EdgeMeta& m) {
        if (m.active) {
            float4 h0 = stage_buf[buf][tid][0];
            float4 h1 = stage_buf[buf][tid][1];
            float4 h2 = stage_buf[buf][tid][2];

            const size_t roff = (size_t)m.row * DD + (size_t)lane16 * 4;
            float* o0 = out + (size_t)0 * N * DD + roff;
            float* o1 = out + (size_t)1 * N * DD + roff;
            float* o2 = out + (size_t)2 * N * DD + roff;

            atomic_fadd(o0 + 0, m.s0 * h0.x);
            atomic_fadd(o0 + 1, m.s0 * h0.y);
            atomic_fadd(o0 + 2, m.s0 * h0.z);
            atomic_fadd(o0 + 3, m.s0 * h0.w);

            atomic_fadd(o1 + 0, m.s1 * h1.x);
            atomic_fadd(o1 + 1, m.s1 * h1.y);
            atomic_fadd(o1 + 2, m.s1 * h1.z);
            atomic_fadd(o1 + 3, m.s1 * h1.w);

            atomic_fadd(o2 + 0, m.s2 * h2.x);
            atomic_fadd(o2 + 1, m.s2 * h2.y);
            atomic_fadd(o2 + 2, m.s2 * h2.z);
            atomic_fadd(o2 + 3, m.s2 * h2.w);
        }
    };

    // ---- 2-deep pipeline: gather stage s+1 into LDS while doing the -------
    // ---- atomic scatter of stage s. ---------------------------------------
    EdgeMeta m_cur = load_meta(0);
    issue_async(0, m_cur);

#pragma unroll 1
    for (int s = 0; s < STAGES; ++s) {
        EdgeMeta m_next;
        if (s + 1 < STAGES) {
            m_next = load_meta(s + 1);
            issue_async((s + 1) & 1, m_next);
            // 3 async ops of stage s+1 may remain in flight; in-order
            // completion => stage s's data is in LDS once cnt <= 3.
            if (__any(m_next.active)) {
                asm volatile("s_wait_asynccnt 0x3" ::: "memory");
            } else {
                asm volatile("s_wait_asynccnt 0x0" ::: "memory");
            }
        } else {
            m_next.active = 0; m_next.row = 0;
            m_next.s0 = m_next.s1 = m_next.s2 = 0.f; m_next.coff = 0;
            asm volatile("s_wait_asynccnt 0x0" ::: "memory");
        }

        consume(s & 1, m_cur);
        m_cur = m_next;
    }
}

// ---------------------------------------------------------------------------
// kernel_launch
//   d_in[0] = H           [C_OUT, N, D]   float32
//   d_in[1] = edge_index  [C_IN, 2, E]    int32
//   d_in[2] = edge_value  [C_IN, E]       float32
//   d_in[3] = weight      [C_OUT, C_IN]   float32
//   d_in[4] = num_nodes   scalar int
//   d_out   = Hs [C_OUT, N, D] ++ filt [C_OUT, C_IN]   (float32)
// ---------------------------------------------------------------------------
extern "C" void kernel_launch(void* const* d_in, const int* in_sizes, int n_in,
                              void* d_out, int out_size, void* d_ws, size_t ws_size,
                              hipStream_t stream) {
    const float* H  = (const float*)d_in[0];
    const int*   EI = (const int*)d_in[1];
    const float* EV = (const float*)d_in[2];
    const float* W  = (const float*)d_in[3];

    const int N = in_sizes[0] / (C_OUT * DD);   // 100000
    const int E = in_sizes[2] / C_IN;           // 250000
    const int M = C_IN * E;                     // 1,000,000

    float* out      = (float*)d_out;
    float* filt_out = out + (size_t)C_OUT * N * DD;  // 12-float tail
    float* filt_ws  = (float*)d_ws;                  // 12 floats of scratch

    // 1) zero the accumulator region (atomics accumulate into it)
    {
        int n4 = (C_OUT * N * DD) / 4;
        int blocks = (n4 + 255) / 256;
        fgt_zero_kernel<<<blocks, 256, 0, stream>>>((float4*)d_out, n4);
    }

    // 2) softmax filter (3x4), written to workspace + output tail
    fgt_softmax_kernel<<<1, 32, 0, stream>>>(W, filt_ws, filt_out);

    // 3) pipelined scatter-add SpMM over all 1M raw edges
    {
        const int edges_per_block = TILE_EDGES * STAGES;   // 160
        int blocks = (M + edges_per_block - 1) / edges_per_block;
        fgt_scatter_kernel<<<blocks, 256, 0, stream>>>(H, EI, EV, filt_ws, out,
                                                       N, E, M);
    }
}